// TimeAttention_New_4707284157161
// MI455X (gfx1250) — compile-verified
//
#include <hip/hip_runtime.h>
#include <stdint.h>

// MI455X / gfx1250, wave32. f32 WMMA path: V_WMMA_F32_16X16X4_F32.
typedef __attribute__((ext_vector_type(2))) float v2f;
typedef __attribute__((ext_vector_type(8))) float v8f;

#define B_ 16
#define L_ 2048
#define D_ 128
#define NEG_FILL (-4294967295.0f)
#define INV_TEMP ((float)(1.0 / (11.313708498984761 + 1e-06)))

#define QSTRIDE 132   // 128 + 4 pad -> bank stride 4 (conflict-free 16-lane column reads)
#define PSTRIDE 68    // 64 + 4 pad

__device__ __forceinline__ v8f wmma_f32(v2f a, v2f b, v8f c) {
  // 8 args: (neg_a, A, neg_b, B, c_mod, C, reuse_a, reuse_b)
  return __builtin_amdgcn_wmma_f32_16x16x4_f32(false, a, false, b, (short)0, c,
                                               false, false);
}

// ballot-based 16-bit row-mask pack: bit r = (mask[q0+r] != 0), r = lane&15.
__device__ __forceinline__ unsigned pack_mask16(const uint8_t* mp) {
  return __builtin_amdgcn_ballot_w32(mp[0] != 0) & 0xffffu;
}

// Async-stage 16 floats (one row segment) of a Q tile into LDS, no VGPR trip.
__device__ __forceinline__ void stage_tile(uint32_t lds_off, uint32_t goff,
                                           const float* base) {
  asm volatile(
      "global_load_async_to_lds_b128 %0, %1, %2\n\t"
      "global_load_async_to_lds_b128 %0, %1, %2 offset:16\n\t"
      "global_load_async_to_lds_b128 %0, %1, %2 offset:32\n\t"
      "global_load_async_to_lds_b128 %0, %1, %2 offset:48" ::"v"(lds_off),
      "v"(goff), "s"(base)
      : "memory");
}

// -------------------------------------------------------------------------
// Kernel 1: per-column softmax stats (softmax is over the q axis!).
// Grid: (L/64, B). Block: 128 threads = 4 waves; wave w owns 16 columns.
// K operand is loop-invariant per lane -> 32 v2f registers from global.
// Q tiles double-buffered via GLOBAL_LOAD_ASYNC_TO_LDS_B128: the stage for
// tile t+1 is issued right after the barrier and waited at the top of the
// next iteration, hiding the whole fetch behind 32 WMMAs. One barrier/tile.
// -------------------------------------------------------------------------
__global__ __launch_bounds__(128) void colstats_kernel(
    const int* __restrict__ T_idx, const float* __restrict__ Dy,
    const uint8_t* __restrict__ mask, const float* __restrict__ time_emb,
    float* __restrict__ colmax, float* __restrict__ colden) {
  const int b = blockIdx.y;
  const int k0 = blockIdx.x * 64;
  const int tid = threadIdx.x;
  const int wave = tid >> 5;
  const int lane = tid & 31;
  const int lmod = lane & 15;
  const int lhi = lane >> 4;            // 0 or 1 (which half of the wave)
  const int db0 = lhi * 2;              // f32 A/B layout: lanes 16-31 hold K=2,3
  const int kbase = k0 + wave * 16;     // this wave's 16 columns

  __shared__ __align__(16) float Qlds[2][16 * QSTRIDE];

  // B operand in registers: lane's column kcol = kbase+lmod, gathered row.
  const int trow = T_idx[b * L_ + kbase + lmod];
  v2f kreg[32];
#pragma unroll
  for (int t = 0; t < 32; ++t)
    kreg[t] = *((const v2f*)&time_emb[(size_t)trow * D_ + 4 * t + db0]);

  // Per-thread staging slot: thread t stages 16 floats of row (t>>3).
  const int srow = tid >> 3, sseg = tid & 7;
  const uint32_t lds_off0 =
      (uint32_t)(uintptr_t)&Qlds[0][srow * QSTRIDE + sseg * 16];
  const uint32_t lds_off1 =
      (uint32_t)(uintptr_t)&Qlds[1][srow * QSTRIDE + sseg * 16];
  const uint32_t gstep = 16 * D_ * sizeof(float);  // bytes per 16-row tile
  uint32_t goff = (uint32_t)(
      (((size_t)(b * L_ + k0 + srow)) * D_ + sseg * 16) * sizeof(float));

  // Prologue: stage the first tile into buffer 0.
  stage_tile(lds_off0, goff, Dy);
  goff += gstep;

  float m_run = -INFINITY;
  float d_run = 0.0f;
  int p = 0;

  for (int q0 = k0; q0 < L_; q0 += 16) {   // causal: q >= k0 for all our cols
    const unsigned mbits = pack_mask16(mask + (b * L_ + q0 + lmod));
    asm volatile("s_wait_asynccnt 0" ::: "memory");  // current tile resident
    __syncthreads();  // ...for everyone; and prev reads of other buffer done
    if (q0 + 16 < L_) {  // overlap next tile's DMA with this tile's WMMAs
      stage_tile(p ? lds_off0 : lds_off1, goff, Dy);
      goff += gstep;
    }
    const float* qtile = Qlds[p];

    // 16(q) x 16(k) score tile: 32 x V_WMMA_F32_16X16X4_F32 over D=128.
    v8f acc = {0, 0, 0, 0, 0, 0, 0, 0};
#pragma unroll
    for (int dd = 0; dd < D_; dd += 4) {
      v2f a = *((const v2f*)&qtile[lmod * QSTRIDE + dd + db0]);
      acc = wmma_f32(a, kreg[dd >> 2], acc);
    }

    // C layout: lane holds column n=lmod, rows m = i + 8*lhi in acc[i].
    const int kcol = kbase + lmod;
    float sc[8];
#pragma unroll
    for (int i = 0; i < 8; ++i) {
      int mrow = i + lhi * 8;
      int q = q0 + mrow;
      bool msk = (kcol > q) || ((mbits >> mrow) & 1u);
      sc[i] = msk ? NEG_FILL : acc[i] * INV_TEMP;
    }
    float tmax = sc[0];
#pragma unroll
    for (int i = 1; i < 8; ++i) tmax = fmaxf(tmax, sc[i]);
    tmax = fmaxf(tmax, __shfl_xor(tmax, 16, 32));  // combine row halves
    float nm = fmaxf(m_run, tmax);
    float psum = 0.0f;
#pragma unroll
    for (int i = 0; i < 8; ++i) psum += __expf(sc[i] - nm);
    psum += __shfl_xor(psum, 16, 32);
    d_run = d_run * __expf(m_run - nm) + psum;
    m_run = nm;
    p ^= 1;
  }

  // q in [0, k0) are all NEG_FILL (causal): fold in analytically. This keeps
  // the fully-masked-column edge case exact (uniform softmax -> denom = L).
  if (k0 > 0) {
    float nm = fmaxf(m_run, NEG_FILL);
    d_run = d_run * __expf(m_run - nm) + (float)k0 * __expf(NEG_FILL - nm);
    m_run = nm;
  }

  if (lane < 16) {
    colmax[b * L_ + kbase + lmod] = m_run;
    colden[b * L_ + kbase + lmod] = d_run;
  }
}

// -------------------------------------------------------------------------
// Kernel 2: att[b,q,:] = sum_k exp(s-m[k])/den[k] * V[k,:].
// Grid: (L/16, B). Block: 128 threads = 4 waves. WG owns a 16-row q tile and
// all 128 output columns (wave w -> d-chunks 2w, 2w+1). k-tiles of 64:
// each wave scores 16 k-columns, P goes through LDS (layout transpose),
// then each wave accumulates P x V into its d-chunks with WMMA.
// Q/A operand is loop-invariant per lane -> 32 v2f registers from global.
// -------------------------------------------------------------------------
__global__ __launch_bounds__(128) void attout_kernel(
    const int* __restrict__ T_idx, const float* __restrict__ Dy,
    const uint8_t* __restrict__ mask, const float* __restrict__ time_emb,
    const float* __restrict__ colmax, const float* __restrict__ colden,
    float* __restrict__ out) {
  const int b = blockIdx.y;
  const int q0 = blockIdx.x * 16;
  const int tid = threadIdx.x;
  const int wave = tid >> 5;
  const int lane = tid & 31;
  const int lmod = lane & 15;
  const int lhi = lane >> 4;
  const int db0 = lhi * 2;

  __shared__ __align__(16) float Plds[16 * PSTRIDE];

  // A operand (Q) in registers: lane's row q0+lmod, whole D, operand layout.
  v2f qreg[32];
#pragma unroll
  for (int t = 0; t < 32; ++t)
    qreg[t] =
        *((const v2f*)&Dy[((size_t)(b * L_ + q0 + lmod)) * D_ + 4 * t + db0]);

  const unsigned mbits = pack_mask16(mask + (b * L_ + q0 + lmod));

  v8f oacc0 = {0, 0, 0, 0, 0, 0, 0, 0};
  v8f oacc1 = {0, 0, 0, 0, 0, 0, 0, 0};

  const int d0 = wave * 32 + lmod;  // this lane's first output column
  const int ntiles = (q0 + 16 + 63) >> 6;  // causal: only k <= q0+15 matter

  for (int kt = 0; kt < ntiles; ++kt) {
    const int kb64 = kt * 64;

    // ---- score phase: this wave's 16 columns, K gathered from global (L2) ----
    const int kcol = kb64 + wave * 16 + lmod;
    const int trow2 = T_idx[b * L_ + kcol];
    v8f sacc = {0, 0, 0, 0, 0, 0, 0, 0};
#pragma unroll
    for (int dd = 0; dd < D_; dd += 4) {
      v2f bm = *((const v2f*)&time_emb[(size_t)trow2 * D_ + dd + db0]);
      sacc = wmma_f32(qreg[dd >> 2], bm, sacc);
    }
    float mk = colmax[b * L_ + kcol];
    float invd = 1.0f / colden[b * L_ + kcol];
#pragma unroll
    for (int i = 0; i < 8; ++i) {
      int mrow = i + lhi * 8;
      int q = q0 + mrow;
      bool msk = (kcol > q) || ((mbits >> mrow) & 1u);
      float s = msk ? NEG_FILL : sacc[i] * INV_TEMP;
      // exp(NEG_FILL - m) is 0 normally, 1/den for fully-masked columns:
      // matches the reference's degenerate uniform softmax exactly.
      Plds[mrow * PSTRIDE + wave * 16 + lmod] = __expf(s - mk) * invd;
    }
    __syncthreads();

    // ---- output phase: oacc += P(16x64) x V(64x16), V direct from global ----
    const float* vbase = Dy + ((size_t)(b * L_ + kb64)) * D_;
#pragma unroll
    for (int kk = 0; kk < 64; kk += 4) {
      int kbi = kk + db0;
      v2f a = *((const v2f*)&Plds[lmod * PSTRIDE + kbi]);
      v2f b0, b1;
      b0.x = vbase[(size_t)kbi * D_ + d0];
      b0.y = vbase[(size_t)(kbi + 1) * D_ + d0];
      b1.x = vbase[(size_t)kbi * D_ + d0 + 16];
      b1.y = vbase[(size_t)(kbi + 1) * D_ + d0 + 16];
      oacc0 = wmma_f32(a, b0, oacc0);
      oacc1 = wmma_f32(a, b1, oacc1);
    }
    __syncthreads();  // protect Plds for next tile
  }

  // ---- tail: columns k beyond the causal range with m[k]==NEG_FILL (fully
  // masked) contribute 1/den[k] * V[k,:] to EVERY row q (uniform softmax). ----
  const int tail0 = ntiles * 64;
  for (int k = tail0; k < L_; ++k) {
    float mk = colmax[b * L_ + k];
    if (mk < -4.0e9f) {
      float wgt = 1.0f / colden[b * L_ + k];
      float v0 = Dy[((size_t)(b * L_ + k)) * D_ + d0];
      float v1 = Dy[((size_t)(b * L_ + k)) * D_ + d0 + 16];
#pragma unroll
      for (int i = 0; i < 8; ++i) {
        oacc0[i] += wgt * v0;
        oacc1[i] += wgt * v1;
      }
    }
  }

  // ---- epilogue: C layout -> global. lane: col d0 (+16), rows i + 8*lhi ----
#pragma unroll
  for (int i = 0; i < 8; ++i) {
    int q = q0 + i + lhi * 8;
    out[((size_t)(b * L_ + q)) * D_ + d0] = oacc0[i];
    out[((size_t)(b * L_ + q)) * D_ + d0 + 16] = oacc1[i];
  }
}

// -------------------------------------------------------------------------
extern "C" void kernel_launch(void* const* d_in, const int* in_sizes, int n_in,
                              void* d_out, int out_size, void* d_ws,
                              size_t ws_size, hipStream_t stream) {
  const int* T_idx = (const int*)d_in[0];
  const float* Dy = (const float*)d_in[1];
  const uint8_t* mask = (const uint8_t*)d_in[2];  // jax bool -> 1 byte
  const float* time_emb = (const float*)d_in[3];
  float* out = (float*)d_out;

  float* colmax = (float*)d_ws;            // B*L f32
  float* colden = colmax + (B_ * L_);      // B*L f32  (256 KB total)

  dim3 g1(L_ / 64, B_), blk(128);
  colstats_kernel<<<g1, blk, 0, stream>>>(T_idx, Dy, mask, time_emb, colmax,
                                          colden);
  dim3 g2(L_ / 16, B_);
  attout_kernel<<<g2, blk, 0, stream>>>(T_idx, Dy, mask, time_emb, colmax,
                                        colden, out);
}